// Brain3DQTUNNetwork_45054206935543
// MI455X (gfx1250) — compile-verified
//
#include <hip/hip_runtime.h>
#include <stdint.h>

// ---------------------------------------------------------------------------
// Brain3DQTUNNetwork: 64^3 neurons, Manhattan radius-2 stencil (24 offsets),
// 50 recurrent steps. Bandwidth-bound.
//   - CSC-style gather map (gmap[k*N+j] = edge id) -> atomic-free SpMV
//   - prev[] staged per-block into LDS with GLOBAL_LOAD_ASYNC_TO_LDS_B32
//     (CDNA5 async-copy path, ASYNCcnt + s_wait_asynccnt)
//   - recurrent working set (~100MB: w, gmap, rows/cols, state) fits MI455X's
//     192MB L2 -> default RT caching; one-shot streams (ext in, spikes out)
//     use non-temporal hints so they don't evict the resident set
//   - edge-parallel STDP kernel (coalesced w r/m/w)
// ---------------------------------------------------------------------------

#define BLK 256
#define NOFF 24

// Linear-index deltas for the 24 Manhattan-radius-2 offsets (dx*4096+dy*64+dz),
// enumerated dx asc, dy asc, dz asc. Only internal consistency between
// build_gmap and step_neuron is required.
__constant__ int c_delta[NOFF] = {
    -8192,                               // (-2, 0, 0)
    -4160, -4097, -4096, -4095, -4032,   // (-1,*)
    -128,  -65,   -64,   -63,            // (0,-2,0) (0,-1,*)
    -2,    -1,     1,     2,             // (0,0,*)
     63,    64,    65,   128,            // (0,1,*) (0,2,0)
     4032,  4095,  4096,  4097,  4160,   // (1,*)
     8192                                // (2, 0, 0)
};

// --- per-call workspace init (harness poisons ws once; we must re-init) -----
__global__ void init_kernel(const float* __restrict__ edge_values,
                            float* __restrict__ w,
                            int*   __restrict__ gmap,
                            float* __restrict__ v,
                            float* __restrict__ prev,
                            int N, int E) {
    int i = blockIdx.x * BLK + threadIdx.x;
    if (i < NOFF * N) gmap[i] = -1;
    if (i < N) { v[i] = -65.0f; prev[i] = 0.0f; }
    if (i < E) w[i] = edge_values[i];
}

// --- invert edge list into dest-major gather map ----------------------------
__global__ void build_gmap(const int* __restrict__ rows,
                           const int* __restrict__ cols,
                           int* __restrict__ gmap,
                           int N, int E) {
    int e = blockIdx.x * BLK + threadIdx.x;
    if (e >= E) return;
    int r = rows[e];
    int c = cols[e];
    int d = c - r;              // unique per offset (bounds-checked edges only)
    int k = -1;
    #pragma unroll
    for (int t = 0; t < NOFF; ++t)
        if (c_delta[t] == d) k = t;
    if (k >= 0) gmap[(size_t)k * N + c] = e;
}

// --- one recurrent step: synaptic gather + QTUN neuron update ---------------
__global__ void __launch_bounds__(BLK)
step_neuron(const float* __restrict__ prev,
            const float* __restrict__ w,
            const int*   __restrict__ gmap,
            const float* __restrict__ ext,      // external_input + t*N
            float* __restrict__ v,
            float* __restrict__ out,
            float* __restrict__ spikes_out,     // d_out + t*N
            int N) {
    __shared__ float sprev[NOFF * BLK];

    const int tid = threadIdx.x;
    const int j   = blockIdx.x * BLK + tid;   // grid is exactly N/BLK blocks

    // Stage the 24 gather windows of prev[] into LDS with async copies
    // (CDNA5 GLOBAL_LOAD_ASYNC_TO_LDS path, tracked by ASYNCcnt).
    #pragma unroll
    for (int k = 0; k < NOFF; ++k) {
        int src = j - c_delta[k];
        src = src < 0 ? 0 : (src >= N ? N - 1 : src);   // clamp; masked by gmap
        unsigned lds_addr = (unsigned)(uintptr_t)&sprev[k * BLK + tid];
        const float* gp = prev + src;
        asm volatile("global_load_async_to_lds_b32 %0, %1, off"
                     :: "v"(lds_addr), "v"(gp)
                     : "memory");
    }
    asm volatile("s_wait_asynccnt 0x0" ::: "memory");
    __syncthreads();

    // Atomic-free segment-sum: syn[j] = sum_k w[gmap[k,j]] * prev[j - delta_k]
    float syn = 0.0f;
    #pragma unroll
    for (int k = 0; k < NOFF; ++k) {
        int e = gmap[(size_t)k * N + j];
        if (e >= 0) syn = fmaf(w[e], sprev[k * BLK + tid], syn);
    }

    const float decay = __expf(-1.0f / 20.0f);          // TAU = 20
    // ext is read exactly once across the whole run: non-temporal load keeps
    // the recurrent working set resident in L2.
    const float I = syn + __builtin_nontemporal_load(&ext[j]);

    float vj = v[j] * decay + I * (1.0f - decay);
    float spike = (vj >= -50.0f) ? 1.0f : 0.0f;         // EXC_THR
    float inhib = (vj <= -70.0f) ? 1.0f : 0.0f;         // INH_THR
    float is_sup = (1.0f - spike) * (1.0f - inhib);
    float sig = 1.0f / (1.0f + __expf(-(vj - (-60.0f)) * 0.5f)); // midpoint=-60
    float o = spike + is_sup * sig;
    vj = vj * (1.0f - spike) + spike * (-65.0f);        // RESET_V

    v[j] = vj;
    out[j] = o;
    // spikes are write-once output: non-temporal store, bypass L2 residency
    __builtin_nontemporal_store(spike, &spikes_out[j]);
}

// --- balanced STDP, edge-parallel (skipped at t==0 by the host loop) --------
// dw = ETA_LTP*pre*post - ETA_LTD*pre*(1-post) - WDECAY*w
//    = pre*(0.015f*post - 0.005f) - 1e-5f*w        (2 FMAs)
__global__ void step_stdp(const int* __restrict__ rows,
                          const int* __restrict__ cols,
                          const float* __restrict__ prev,   // carry-in "prev"
                          const float* __restrict__ out,    // this step's out
                          float* __restrict__ w,
                          int E) {
    int e = blockIdx.x * BLK + threadIdx.x;
    if (e >= E) return;
    float pre  = prev[cols[e]];
    float post = out[rows[e]];
    float wv   = w[e];
    float wvn  = fmaf(pre, fmaf(0.015f, post, -0.005f), fmaf(-1e-5f, wv, wv));
    wvn = wvn < 0.0f ? 0.0f : (wvn > 1.0f ? 1.0f : wvn);
    w[e] = wvn;
}

extern "C" void kernel_launch(void* const* d_in, const int* in_sizes, int n_in,
                              void* d_out, int out_size, void* d_ws, size_t ws_size,
                              hipStream_t stream) {
    const float* ext  = (const float*)d_in[0];  // (steps, N)
    const float* ev   = (const float*)d_in[1];  // (E,)
    const int*   rows = (const int*)  d_in[2];  // (E,)
    const int*   cols = (const int*)  d_in[3];  // (E,)

    const int N     = 64 * 64 * 64;
    const int steps = in_sizes[0] / N;
    const int E     = in_sizes[1];

    // Workspace layout (256B aligned slices)
    char* ws = (char*)d_ws;
    auto align256 = [](size_t x) { return (x + 255) & ~(size_t)255; };
    float* w    = (float*)ws;                 ws += align256((size_t)E * 4);
    int*   gmap = (int*)ws;                   ws += align256((size_t)NOFF * N * 4);
    float* v    = (float*)ws;                 ws += align256((size_t)N * 4);
    float* bufA = (float*)ws;                 ws += align256((size_t)N * 4);
    float* bufB = (float*)ws;

    const int initGrid = (NOFF * N + BLK - 1) / BLK;
    init_kernel<<<initGrid, BLK, 0, stream>>>(ev, w, gmap, v, bufA, N, E);

    const int edgeGrid = (E + BLK - 1) / BLK;
    build_gmap<<<edgeGrid, BLK, 0, stream>>>(rows, cols, gmap, N, E);

    float* prev = bufA;
    float* out  = bufB;
    const int neuronGrid = N / BLK;

    for (int t = 0; t < steps; ++t) {
        step_neuron<<<neuronGrid, BLK, 0, stream>>>(
            prev, w, gmap, ext + (size_t)t * N,
            v, out, (float*)d_out + (size_t)t * N, N);
        if (t > 0) {
            step_stdp<<<edgeGrid, BLK, 0, stream>>>(rows, cols, prev, out, w, E);
        }
        float* tmp = prev; prev = out; out = tmp;
    }
}